// MoEGate_36283883716950
// MI455X (gfx1250) — compile-verified
//
#include <hip/hip_runtime.h>
#include <hip/hip_bf16.h>

typedef float v2f __attribute__((ext_vector_type(2)));
typedef float v4f __attribute__((ext_vector_type(4)));
typedef float v8f __attribute__((ext_vector_type(8)));

#define HID    7168
#define NEXP   256
#define NTOK   16384
#define TOPK   8
#define BLK_M  32   // tokens per block

// Fused MoE gate: fp32 WMMA GEMM (logits) + softmax + top-8 + norm.
// Block = 256 threads = 8 waves. Wave w owns experts [32w, 32w+32) (2 N-tiles);
// block owns 32 tokens (2 M-tiles). K advances 8/iter via one b128 load per
// operand: lanes 0-15 carry physical cols k..k+3, lanes 16-31 carry k+4..k+7;
// (.x,.y) and (.z,.w) form two K=4 WMMA fragments under a consistent logical-K
// permutation of A and B (sum over K is order-independent).
__global__ __launch_bounds__(256) void moe_gate_kernel(
    const float* __restrict__ A,    // hidden_states [NTOK, HID]
    const float* __restrict__ W,    // weight        [NEXP, HID]
    int*   __restrict__ out_idx,    // [NTOK, TOPK]
    float* __restrict__ out_w,      // [NTOK, TOPK]
    int*   __restrict__ out_row)    // [NTOK, TOPK]
{
    __shared__ float lds[BLK_M][NEXP + 1];   // row stride 257 -> conflict-free

    const int tid  = threadIdx.x;
    const int lane = tid & 31;
    const int wav  = tid >> 5;        // 0..7
    const int lmod = lane & 15;       // row/col within 16-wide tile
    const int lhi  = lane >> 4;       // 0: K slots {0,1}; 1: K slots {2,3}
    const int m0   = blockIdx.x * BLK_M;
    const int n0   = wav * 32;

    const float* a0 = A + (size_t)(m0 + lmod) * HID + lhi * 4;
    const float* a1 = a0 + (size_t)16 * HID;
    const float* b0 = W + (size_t)(n0 + lmod) * HID + lhi * 4;
    const float* b1 = b0 + (size_t)16 * HID;

    v8f acc00 = {}, acc01 = {}, acc10 = {}, acc11 = {};

#pragma unroll 2
    for (int k = 0; k < HID; k += 8) {
        v4f A0 = *(const v4f*)(a0 + k);
        v4f A1 = *(const v4f*)(a1 + k);
        v4f B0 = *(const v4f*)(b0 + k);
        v4f B1 = *(const v4f*)(b1 + k);

        v2f a0l = {A0.x, A0.y}, a0h = {A0.z, A0.w};
        v2f a1l = {A1.x, A1.y}, a1h = {A1.z, A1.w};
        v2f b0l = {B0.x, B0.y}, b0h = {B0.z, B0.w};
        v2f b1l = {B1.x, B1.y}, b1h = {B1.z, B1.w};

        // 8 args: (neg_a, A, neg_b, B, c_mod, C, reuse_a, reuse_b)
        acc00 = __builtin_amdgcn_wmma_f32_16x16x4_f32(false, a0l, false, b0l, (short)0, acc00, false, false);
        acc01 = __builtin_amdgcn_wmma_f32_16x16x4_f32(false, a0l, false, b1l, (short)0, acc01, false, false);
        acc10 = __builtin_amdgcn_wmma_f32_16x16x4_f32(false, a1l, false, b0l, (short)0, acc10, false, false);
        acc11 = __builtin_amdgcn_wmma_f32_16x16x4_f32(false, a1l, false, b1l, (short)0, acc11, false, false);
        acc00 = __builtin_amdgcn_wmma_f32_16x16x4_f32(false, a0h, false, b0h, (short)0, acc00, false, false);
        acc01 = __builtin_amdgcn_wmma_f32_16x16x4_f32(false, a0h, false, b1h, (short)0, acc01, false, false);
        acc10 = __builtin_amdgcn_wmma_f32_16x16x4_f32(false, a1h, false, b0h, (short)0, acc10, false, false);
        acc11 = __builtin_amdgcn_wmma_f32_16x16x4_f32(false, a1h, false, b1h, (short)0, acc11, false, false);
    }

    // C layout: VGPR r, lane L -> row = r + 8*(L/16), col = L%16. Scatter to LDS.
#pragma unroll
    for (int r = 0; r < 8; ++r) {
        const int trow = r + 8 * lhi;
        const int ncol = n0 + lmod;
        lds[trow     ][ncol     ] = acc00[r];
        lds[trow     ][ncol + 16] = acc01[r];
        lds[trow + 16][ncol     ] = acc10[r];
        lds[trow + 16][ncol + 16] = acc11[r];
    }
    __syncthreads();

    // Epilogue: one thread per token (32 active); negligible vs 117 MFLOP of GEMM/block.
    if (tid < BLK_M) {
        const int t    = tid;
        const int gtok = m0 + t;

        float mx = -__builtin_inff();
        for (int j = 0; j < NEXP; ++j) mx = fmaxf(mx, lds[t][j]);
        float Z = 0.0f;
        for (int j = 0; j < NEXP; ++j) Z += __expf(lds[t][j] - mx);

        float wv[TOPK];
        int   wi[TOPK];
        float s8 = 0.0f;
#pragma unroll
        for (int k = 0; k < TOPK; ++k) {
            float bv = -__builtin_inff();
            int   bi = 0;
            for (int j = 0; j < NEXP; ++j) {
                float v = lds[t][j];
                if (v > bv) { bv = v; bi = j; }   // strict > : lowest index wins ties
            }
            lds[t][bi] = -__builtin_inff();
            float p = __expf(bv - mx) / Z;
            wv[k] = p; wi[k] = bi; s8 += p;
        }
        const float inv = 1.0f / (s8 + 1e-20f);
#pragma unroll
        for (int k = 0; k < TOPK; ++k) {
            out_idx[gtok * TOPK + k] = wi[k];
            out_w  [gtok * TOPK + k] = wv[k] * inv;
            out_row[gtok * TOPK + k] = k * NTOK + gtok;   // arange(T*K).reshape(K,T).T
        }
    }
}

extern "C" void kernel_launch(void* const* d_in, const int* in_sizes, int n_in,
                              void* d_out, int out_size, void* d_ws, size_t ws_size,
                              hipStream_t stream) {
    (void)in_sizes; (void)n_in; (void)out_size; (void)d_ws; (void)ws_size;
    const float* hs = (const float*)d_in[0];   // hidden_states [16384,7168] f32
    const float* wt = (const float*)d_in[1];   // weight        [256,7168]  f32
    // d_out = topk_idx (int32) ++ topk_weight (f32) ++ row_idx (int32), flat.
    int*   out_idx = (int*)d_out;
    float* out_w   = (float*)d_out + (size_t)NTOK * TOPK;
    int*   out_row = (int*)d_out   + (size_t)2 * NTOK * TOPK;

    dim3 grid(NTOK / BLK_M);   // 512 blocks
    dim3 block(256);           // 8 wave32 waves
    moe_gate_kernel<<<grid, block, 0, stream>>>(hs, wt, out_idx, out_w, out_row);
}